// GATModule_50663434224135
// MI455X (gfx1250) — compile-verified
//
#include <hip/hip_runtime.h>

#define N_NODES 100000
#define N_EDGES 1600000
#define DIM 128
#define HEADS 2
#define HD 256            // HEADS * DIM
#define NEG_SLOPE 0.2f
#define SEG_EPS 1e-16f

typedef __attribute__((ext_vector_type(2))) float v2f;
typedef __attribute__((ext_vector_type(8))) float v8f;

__device__ __forceinline__ float wave_sum(float v) {
#pragma unroll
  for (int off = 16; off > 0; off >>= 1) v += __shfl_xor(v, off, 32);
  return v;
}

// ---------------- LayerNorm + ReLU: one wave32 per node ----------------
__global__ void ln_relu_kernel(const float* __restrict__ x,
                               const float* __restrict__ gamma,
                               const float* __restrict__ beta,
                               float* __restrict__ h) {
  const int wv = threadIdx.x >> 5, lane = threadIdx.x & 31;
  const int node = blockIdx.x * 8 + wv;
  if (node >= N_NODES) return;
  float4 v = ((const float4*)(x + (size_t)node * DIM))[lane];
  float mu = wave_sum(v.x + v.y + v.z + v.w) * (1.0f / DIM);
  float4 d = make_float4(v.x - mu, v.y - mu, v.z - mu, v.w - mu);
  float var = wave_sum(d.x * d.x + d.y * d.y + d.z * d.z + d.w * d.w) * (1.0f / DIM);
  float inv = rsqrtf(var + 1e-5f);
  float4 g = ((const float4*)gamma)[lane];
  float4 b = ((const float4*)beta)[lane];
  float4 o;
  o.x = fmaxf(d.x * inv * g.x + b.x, 0.0f);
  o.y = fmaxf(d.y * inv * g.y + b.y, 0.0f);
  o.z = fmaxf(d.z * inv * g.z + b.z, 0.0f);
  o.w = fmaxf(d.w * inv * g.w + b.w, 0.0f);
  ((float4*)(h + (size_t)node * DIM))[lane] = o;
}

// ---------------- Fused WMMA f32 GEMM: xl, xr, and out-base ----------------
// blockIdx.y: 0,1 -> Wl cols [y*128,+128); 2,3 -> Wr; 4 -> res_W (128 cols),
// res path writes d_out = x + h@res_W + bias (accumulation base).
__global__ void gemm_wmma_kernel(const float* __restrict__ h,
                                 const float* __restrict__ Wl, const float* __restrict__ bl,
                                 const float* __restrict__ Wr, const float* __restrict__ br,
                                 const float* __restrict__ resW, const float* __restrict__ bias,
                                 const float* __restrict__ x,
                                 float* __restrict__ xl, float* __restrict__ xr,
                                 float* __restrict__ outBase) {
  __shared__ float As[16][DIM + 2];   // +2 pad: A-frag column reads hit distinct banks
  const int i0 = blockIdx.x * 16;
  const int seg = blockIdx.y;

  for (int idx = threadIdx.x; idx < 16 * DIM; idx += 256) {
    int r = idx >> 7, c = idx & (DIM - 1);
    As[r][c] = h[(size_t)(i0 + r) * DIM + c];
  }
  __syncthreads();

  const int wv = threadIdx.x >> 5, lane = threadIdx.x & 31;
  const int half = lane >> 4, m = lane & 15;

  const float* W; int ldw, colbase;
  if (seg < 2)      { W = Wl;   ldw = HD;  colbase = seg * 128 + wv * 16; }
  else if (seg < 4) { W = Wr;   ldw = HD;  colbase = (seg - 2) * 128 + wv * 16; }
  else              { W = resW; ldw = DIM; colbase = wv * 16; }

  v8f acc = {0.f, 0.f, 0.f, 0.f, 0.f, 0.f, 0.f, 0.f};
#pragma unroll 8
  for (int kk = 0; kk < DIM / 4; ++kk) {
    const int k = kk * 4 + 2 * half;      // lanes 16-31 handle K+2,K+3 (ISA A 16x4 layout)
    v2f a, b;
    a.x = As[m][k];
    a.y = As[m][k + 1];
    const float* wp = W + (size_t)k * ldw + colbase + m;
    b.x = wp[0];
    b.y = wp[ldw];
    acc = __builtin_amdgcn_wmma_f32_16x16x4_f32(false, a, false, b, (short)0, acc,
                                                false, false);
  }

  const int col = colbase + m;
  if (seg < 4) {
    const float bb = (seg < 2) ? bl[col] : br[col];
    float* dst = (seg < 2) ? xl : xr;
#pragma unroll
    for (int v = 0; v < 8; ++v) {
      int row = i0 + v + 8 * half;        // C/D layout: VGPR v -> M = v + 8*half
      dst[(size_t)row * HD + col] = acc[v] + bb;
    }
  } else {
    const float bb = bias[col];
#pragma unroll
    for (int v = 0; v < 8; ++v) {
      int row = i0 + v + 8 * half;
      outBase[(size_t)row * DIM + col] = acc[v] + bb + x[(size_t)row * DIM + col];
    }
  }
}

// ---------------- segment softmax state init ----------------
__global__ void init_mz_kernel(unsigned* __restrict__ mEnc, float* __restrict__ z) {
  int i = blockIdx.x * 256 + threadIdx.x;
  if (i < N_NODES * HEADS) { mEnc[i] = 0u; z[i] = 0.0f; }
}

// ---------------- P1: attention logits + segment max (wave per edge) ----------------
__global__ void edge_alpha_kernel(const int* __restrict__ ei,
                                  const float* __restrict__ xl,
                                  const float* __restrict__ xr,
                                  const float* __restrict__ att,
                                  float* __restrict__ alpha,
                                  unsigned* __restrict__ mEnc) {
  const int wv = threadIdx.x >> 5, lane = threadIdx.x & 31;
  const int e = blockIdx.x * 8 + wv;
  if (e >= N_EDGES) return;
  const int src = ei[e];
  const int dst = ei[N_EDGES + e];
#pragma unroll
  for (int hh = 0; hh < HEADS; ++hh) {
    float4 a = ((const float4*)(xl + (size_t)src * HD + hh * DIM))[lane];
    float4 b = ((const float4*)(xr + (size_t)dst * HD + hh * DIM))[lane];
    float4 t = ((const float4*)(att + hh * DIM))[lane];
    float4 s = make_float4(a.x + b.x, a.y + b.y, a.z + b.z, a.w + b.w);
    s.x = s.x > 0.0f ? s.x : NEG_SLOPE * s.x;
    s.y = s.y > 0.0f ? s.y : NEG_SLOPE * s.y;
    s.z = s.z > 0.0f ? s.z : NEG_SLOPE * s.z;
    s.w = s.w > 0.0f ? s.w : NEG_SLOPE * s.w;
    float p = wave_sum(s.x * t.x + s.y * t.y + s.z * t.z + s.w * t.w);
    if (lane == 0) {
      alpha[e * HEADS + hh] = p;
      unsigned u = __float_as_uint(p);
      u = (u & 0x80000000u) ? ~u : (u | 0x80000000u);  // monotonic float->uint
      atomicMax(&mEnc[dst * HEADS + hh], u);
    }
  }
}

// ---------------- P2: exp(alpha - max) + segment sum (thread per edge-head) ----------------
__global__ void edge_exp_kernel(const int* __restrict__ ei,
                                float* __restrict__ alpha,
                                const unsigned* __restrict__ mEnc,
                                float* __restrict__ z) {
  int t = blockIdx.x * 256 + threadIdx.x;
  if (t >= N_EDGES * HEADS) return;
  int e = t >> 1, hh = t & 1;
  int dst = ei[N_EDGES + e];
  unsigned u = mEnc[dst * HEADS + hh];
  float mval = (u & 0x80000000u) ? __uint_as_float(u ^ 0x80000000u)
                                 : __uint_as_float(~u);
  float a = expf(alpha[t] - mval);
  alpha[t] = a;
  atomicAdd(&z[dst * HEADS + hh], a);
}

// ---------------- P3: weighted scatter, head-mean folded into weights ----------------
__global__ void edge_scatter_kernel(const int* __restrict__ ei,
                                    const float* __restrict__ xl,
                                    const float* __restrict__ alpha,
                                    const float* __restrict__ z,
                                    float* __restrict__ out) {
  const int wv = threadIdx.x >> 5, lane = threadIdx.x & 31;
  const int e = blockIdx.x * 8 + wv;
  if (e >= N_EDGES) return;
  const int src = ei[e];
  const int dst = ei[N_EDGES + e];
  const float w0 = alpha[e * 2 + 0] / (z[dst * 2 + 0] + SEG_EPS) * (1.0f / HEADS);
  const float w1 = alpha[e * 2 + 1] / (z[dst * 2 + 1] + SEG_EPS) * (1.0f / HEADS);
  float4 a = ((const float4*)(xl + (size_t)src * HD))[lane];        // head 0
  float4 b = ((const float4*)(xl + (size_t)src * HD + DIM))[lane];  // head 1
  float4 c = make_float4(w0 * a.x + w1 * b.x, w0 * a.y + w1 * b.y,
                         w0 * a.z + w1 * b.z, w0 * a.w + w1 * b.w);
  float* o = out + (size_t)dst * DIM + lane * 4;
  atomicAdd(o + 0, c.x);
  atomicAdd(o + 1, c.y);
  atomicAdd(o + 2, c.z);
  atomicAdd(o + 3, c.w);
}

extern "C" void kernel_launch(void* const* d_in, const int* in_sizes, int n_in,
                              void* d_out, int out_size, void* d_ws, size_t ws_size,
                              hipStream_t stream) {
  const float* x     = (const float*)d_in[0];
  const int*   ei    = (const int*)d_in[1];
  const float* gamma = (const float*)d_in[2];
  const float* beta  = (const float*)d_in[3];
  const float* Wl    = (const float*)d_in[4];
  const float* bl    = (const float*)d_in[5];
  const float* Wr    = (const float*)d_in[6];
  const float* br    = (const float*)d_in[7];
  const float* att   = (const float*)d_in[8];
  const float* resW  = (const float*)d_in[9];
  const float* bias  = (const float*)d_in[10];

  char* ws = (char*)d_ws;
  float*    h     = (float*)ws;    ws += (size_t)N_NODES * DIM * sizeof(float);
  float*    xl    = (float*)ws;    ws += (size_t)N_NODES * HD * sizeof(float);
  float*    xr    = (float*)ws;    ws += (size_t)N_NODES * HD * sizeof(float);
  float*    alpha = (float*)ws;    ws += (size_t)N_EDGES * HEADS * sizeof(float);
  unsigned* mEnc  = (unsigned*)ws; ws += (size_t)N_NODES * HEADS * sizeof(unsigned);
  float*    z     = (float*)ws;
  float*    out   = (float*)d_out;

  ln_relu_kernel<<<N_NODES / 8, 256, 0, stream>>>(x, gamma, beta, h);

  dim3 g(N_NODES / 16, 5);
  gemm_wmma_kernel<<<g, 256, 0, stream>>>(h, Wl, bl, Wr, br, resW, bias, x, xl, xr, out);

  init_mz_kernel<<<(N_NODES * HEADS + 255) / 256, 256, 0, stream>>>(mEnc, z);
  edge_alpha_kernel<<<N_EDGES / 8, 256, 0, stream>>>(ei, xl, xr, att, alpha, mEnc);
  edge_exp_kernel<<<(N_EDGES * HEADS) / 256, 256, 0, stream>>>(ei, alpha, mEnc, z);
  edge_scatter_kernel<<<N_EDGES / 8, 256, 0, stream>>>(ei, xl, alpha, z, out);
}